// SimpleMLP_31310311588408
// MI455X (gfx1250) — compile-verified
//
#include <hip/hip_runtime.h>
#include <hip/hip_bf16.h>
#include <math.h>

#define HDIM     256
#define SEGS     4096

typedef __attribute__((ext_vector_type(2))) float v2f;
typedef __attribute__((ext_vector_type(8))) float v8f;

// ---------------------------------------------------------------------------
// Zero-fill workspace (graph-capture-safe; no hipMemset needed)
// ---------------------------------------------------------------------------
__global__ void zero_f32_kernel(float* __restrict__ p, int n) {
    int i = blockIdx.x * blockDim.x + threadIdx.x;
    if (i < n) p[i] = 0.0f;
}

// ---------------------------------------------------------------------------
// Segment-sum + counts.  batch is sorted, so each block accumulates locally
// in registers and flushes with global f32 atomics only at segment boundaries.
// Block = 256 threads = 4 row-groups x 64 column-lanes; each lane owns 4
// columns (float4 loads -> 1 KB coalesced per row per group).
// ---------------------------------------------------------------------------
__global__ void segsum_kernel(const float* __restrict__ x,
                              const int*   __restrict__ batch,
                              float*       __restrict__ sums,
                              float*       __restrict__ cnt,
                              int rowsPerBlock) {
    const int t    = threadIdx.x;
    const int grp  = t >> 6;        // 0..3  (row group; 2 waves each -> wave-uniform seg)
    const int lane = t & 63;        // 0..63 (column lane)
    const int c    = lane << 2;     // column base (4 floats)
    const long r0  = (long)blockIdx.x * rowsPerBlock;

    float4 acc = make_float4(0.f, 0.f, 0.f, 0.f);
    float  localCnt = 0.f;
    int    curSeg = -1;

    for (int i = grp; i < rowsPerBlock; i += 4) {
        const long row = r0 + i;
        const int  seg = batch[row];            // wave-uniform within a group
        if (seg != curSeg) {
            if (curSeg >= 0) {
                float* s = sums + (size_t)curSeg * HDIM + c;
                atomicAdd(s + 0, acc.x);
                atomicAdd(s + 1, acc.y);
                atomicAdd(s + 2, acc.z);
                atomicAdd(s + 3, acc.w);
                if (lane == 0) atomicAdd(cnt + curSeg, localCnt);
            }
            acc = make_float4(0.f, 0.f, 0.f, 0.f);
            localCnt = 0.f;
            curSeg = seg;
        }
        const float4 v = *reinterpret_cast<const float4*>(x + row * HDIM + c);
        acc.x += v.x; acc.y += v.y; acc.z += v.z; acc.w += v.w;
        localCnt += 1.f;
    }
    if (curSeg >= 0) {
        float* s = sums + (size_t)curSeg * HDIM + c;
        atomicAdd(s + 0, acc.x);
        atomicAdd(s + 1, acc.y);
        atomicAdd(s + 2, acc.z);
        atomicAdd(s + 3, acc.w);
        if (lane == 0) atomicAdd(cnt + curSeg, localCnt);
    }
}

// ---------------------------------------------------------------------------
// pooled = sums / max(cnt, 1)   (in place; one block per segment)
// ---------------------------------------------------------------------------
__global__ void seg_mean_kernel(float* __restrict__ sums,
                                const float* __restrict__ cnt) {
    const int s = blockIdx.x;
    const int t = threadIdx.x;
    const float inv = 1.0f / fmaxf(cnt[s], 1.0f);
    sums[(size_t)s * HDIM + t] *= inv;
}

// ---------------------------------------------------------------------------
// out[M x 256] = act(A[M x 256] @ W[256 x 256] + bias)
// One wave per 16x16 C tile using V_WMMA_F32_16X16X4_F32 (full f32 accum).
//   A frag (16x4, 2 VGPR): lanes 0-15 -> K k+0,k+1 ; lanes 16-31 -> K k+2,k+3
//   B frag (4x16, 2 VGPR): VGPR0 = row k(+2), VGPR1 = row k+1(+3), N = lane&15
//   C/D   (16x16, 8 VGPR): VGPR r -> M = r + 8*(lane>=16), N = lane&15
// ---------------------------------------------------------------------------
__global__ void wmma_gemm_kernel(const float* __restrict__ A,
                                 const float* __restrict__ W,
                                 const float* __restrict__ bias,
                                 float*       __restrict__ out,
                                 int M, int doGelu) {
    const int wavesPerBlock = blockDim.x >> 5;
    const int waveId = blockIdx.x * wavesPerBlock + (threadIdx.x >> 5);
    const int tilesN = HDIM / 16;              // 16
    const int tm = waveId / tilesN;            // tile row  (M/16 of them)
    const int tn = waveId % tilesN;            // tile col
    const int lane = threadIdx.x & 31;
    const int half = lane >> 4;                // 0: K+0/+1, 1: K+2/+3
    const int l    = lane & 15;

    // A addressing: row = tm*16 + l, k-offset = half*2
    const float* arow = A + (size_t)(tm * 16 + l) * HDIM + half * 2;
    // B addressing: col = tn*16 + l, rows k + half*2 and k + half*2 + 1
    const float* bcol = W + (size_t)(half * 2) * HDIM + tn * 16 + l;

    v8f c = {};
#pragma unroll 4
    for (int k = 0; k < HDIM; k += 4) {
        const float2 av = *reinterpret_cast<const float2*>(arow + k);
        v2f a, b;
        a.x = av.x;
        a.y = av.y;
        b.x = bcol[(size_t)k * HDIM];
        b.y = bcol[(size_t)k * HDIM + HDIM];
        // D = A(16x4) * B(4x16) + C   -> v_wmma_f32_16x16x4_f32
        c = __builtin_amdgcn_wmma_f32_16x16x4_f32(
                /*neg_a=*/false, a, /*neg_b=*/false, b,
                /*c_mod=*/(short)0, c, /*reuse_a=*/false, /*reuse_b=*/false);
    }

    const int n  = tn * 16 + l;
    const float bv = bias[n];
#pragma unroll
    for (int r = 0; r < 8; ++r) {
        const int m = tm * 16 + r + half * 8;
        float v = c[r] + bv;
        if (doGelu) {
            // exact GELU: 0.5*x*(1+erf(x/sqrt(2)))
            v = 0.5f * v * (1.0f + erff(v * 0.70710678118654752440f));
        }
        out[(size_t)m * HDIM + n] = v;
    }
}

// ---------------------------------------------------------------------------
// Inputs (setup_inputs order):
//   0: x [N,256] f32   1: edge_index (unused)   2: edge_type (unused)
//   3: batch [N] i32   4: W1 [256,256] f32      5: b1 [256] f32
//   6: W2 [256,256]    7: b2 [256]
// Output: [4096, 256] f32
// ---------------------------------------------------------------------------
extern "C" void kernel_launch(void* const* d_in, const int* in_sizes, int n_in,
                              void* d_out, int out_size, void* d_ws, size_t ws_size,
                              hipStream_t stream) {
    (void)n_in; (void)out_size; (void)ws_size;

    const float* x     = (const float*)d_in[0];
    const int*   batch = (const int*)  d_in[3];
    const float* W1    = (const float*)d_in[4];
    const float* b1    = (const float*)d_in[5];
    const float* W2    = (const float*)d_in[6];
    const float* b2    = (const float*)d_in[7];
    float*       out   = (float*)d_out;

    const int N = in_sizes[3];                 // 1048576 rows

    // Workspace layout: sums[SEGS*HDIM] | cnt[SEGS] | h[SEGS*HDIM]
    float* sums = (float*)d_ws;
    float* cnt  = sums + (size_t)SEGS * HDIM;
    float* h    = cnt + SEGS;

    // 1) zero accumulators
    const int nz = SEGS * HDIM + SEGS;
    zero_f32_kernel<<<(nz + 255) / 256, 256, 0, stream>>>(sums, nz);

    // 2) segment sums + counts (256 rows per block)
    const int rowsPerBlock = 256;
    segsum_kernel<<<N / rowsPerBlock, 256, 0, stream>>>(x, batch, sums, cnt, rowsPerBlock);

    // 3) mean (in place: sums -> pooled)
    seg_mean_kernel<<<SEGS, HDIM, 0, stream>>>(sums, cnt);

    // 4) h = gelu(pooled @ W1 + b1)   — WMMA f32
    const int tiles = (SEGS / 16) * (HDIM / 16);   // 4096 waves
    const int wavesPerBlock = 8;                   // 256 threads
    wmma_gemm_kernel<<<tiles / wavesPerBlock, wavesPerBlock * 32, 0, stream>>>(
        sums, W1, b1, h, SEGS, /*doGelu=*/1);

    // 5) out = h @ W2 + b2            — WMMA f32
    wmma_gemm_kernel<<<tiles / wavesPerBlock, wavesPerBlock * 32, 0, stream>>>(
        h, W2, b2, out, SEGS, /*doGelu=*/0);
}